// TwoPhaseBlockAttnRes_69209103008152
// MI455X (gfx1250) — compile-verified
//
#include <hip/hip_runtime.h>
#include <math.h>

typedef float v2f __attribute__((ext_vector_type(2)));
typedef float v8f __attribute__((ext_vector_type(8)));

namespace {
constexpr int S  = 8;
constexpr int N  = 8;
constexpr int B  = 2;
constexpr int T  = 2048;
constexpr int D  = 1024;
constexpr int TT = 4;              // t-values per workgroup
constexpr int VP = 1040;           // LDS row pitch (floats), padded to dodge bank conflicts
constexpr int RSTRIDE = B * T * D; // element stride between rows (n or s index): 4M floats
constexpr float EPSI  = 1e-6f;
constexpr float SCALE = 0.03125f;  // 1/sqrt(1024)
}

__global__ __launch_bounds__(256)
void twophase_attn_kernel(const float* __restrict__ q,     // [S,D]
                          const float* __restrict__ vrep,  // [N,B,T,D]
                          const float* __restrict__ psum,  // [S,B,T,D]
                          const float* __restrict__ wn,    // [D]
                          float* __restrict__ out)         // merged_out | merged_max | merged_lse
{
    __shared__ float sV[N * VP];   // V rows for current (b,t)
    __shared__ float sDot[S * N];  // raw q·(w∘V) dots: [s][n]
    __shared__ float sSSV[N];
    __shared__ float sSSP[S];
    __shared__ float sDQP[S];
    __shared__ float sCA[S * N];   // coefA[s][n]
    __shared__ float sCP[S];       // coefP[s]

    const int tid  = threadIdx.x;
    const int wave = tid >> 5;
    const int lane = tid & 31;

    const int b  = blockIdx.x / (T / TT);
    const int t0 = (blockIdx.x % (T / TT)) * TT;

    float* outMax = out + (long)S * B * T * D;
    float* outLse = outMax + (long)S * B * T;

    for (int ti = 0; ti < TT; ++ti) {
        const int t = t0 + ti;
        // Uniform (SGPR) base pointers for this (b,t).
        const float* vB = vrep + ((long)b * T + t) * D;
        const float* pB = psum + ((long)b * T + t) * D;
        float*       oB = out  + ((long)b * T + t) * D;

        // Per-row uniform bases, pinned into SGPRs so the optimizer cannot re-fold
        // them into base + 16MB-immediate 64-bit VALU chains. Each access then
        // lowers to saddr + 32-bit voffset (GVS mode, scale_offset).
        const float* vRow[N];
        const float* pRow[S];
        float*       oRow[S];
        #pragma unroll
        for (int r = 0; r < N; ++r) {
            const float* vp = vB + r * RSTRIDE;
            const float* pp = pB + r * RSTRIDE;
            float*       op = oB + r * RSTRIDE;
            asm("" : "+s"(vp));
            asm("" : "+s"(pp));
            asm("" : "+s"(op));
            vRow[r] = vp; pRow[r] = pp; oRow[r] = op;
        }

        __syncthreads();  // sV reuse across t iterations

        // ---- Phase A: stage V rows into LDS (coalesced float4), prefetch next t ----
        {
            const int c = tid * 4;
            #pragma unroll
            for (int r = 0; r < N; ++r) {
                const float4 vv = *(const float4*)(vRow[r] + c);
                *(float4*)&sV[r * VP + c] = vv;
            }
            if (ti + 1 < TT) {
                const int pr  = tid >> 5;
                const int off = D + (tid & 31) * 32;  // next t, 128B granules
                __builtin_prefetch(vRow[pr] + off, 0, 1);
                __builtin_prefetch(pRow[pr] + off, 0, 1);
            }
        }
        __syncthreads();

        // ---- Phase B: wave w owns V row w and P row w; lane-strided float4 reductions ----
        {
            float dv[S] = {0.f,0.f,0.f,0.f,0.f,0.f,0.f,0.f};
            float ssv = 0.f, ssp = 0.f, dqp = 0.f;
            const float* vrow = &sV[wave * VP];
            const int    poff = wave * RSTRIDE;
            #pragma unroll
            for (int j = 0; j < 8; ++j) {
                const int d = j * 128 + lane * 4;
                const float4 v4 = *(const float4*)(vrow + d);
                const float4 p4 = *(const float4*)(pB + (poff + d));
                const float4 w4 = *(const float4*)(wn + d);
                ssv += v4.x*v4.x + v4.y*v4.y + v4.z*v4.z + v4.w*v4.w;
                ssp += p4.x*p4.x + p4.y*p4.y + p4.z*p4.z + p4.w*p4.w;
                float4 vw; vw.x = v4.x*w4.x; vw.y = v4.y*w4.y; vw.z = v4.z*w4.z; vw.w = v4.w*w4.w;
                float4 pw; pw.x = p4.x*w4.x; pw.y = p4.y*w4.y; pw.z = p4.z*w4.z; pw.w = p4.w*w4.w;
                #pragma unroll
                for (int s = 0; s < S; ++s) {
                    const float4 q4 = *(const float4*)(q + s * D + d);
                    dv[s] += q4.x*vw.x + q4.y*vw.y + q4.z*vw.z + q4.w*vw.w;
                    if (s == wave)
                        dqp += q4.x*pw.x + q4.y*pw.y + q4.z*pw.z + q4.w*pw.w;
                }
            }
            #pragma unroll
            for (int off = 16; off > 0; off >>= 1) {
                ssv += __shfl_xor(ssv, off, 32);
                ssp += __shfl_xor(ssp, off, 32);
                dqp += __shfl_xor(dqp, off, 32);
                #pragma unroll
                for (int s = 0; s < S; ++s) dv[s] += __shfl_xor(dv[s], off, 32);
            }
            if (lane == 0) {
                sSSV[wave] = ssv; sSSP[wave] = ssp; sDQP[wave] = dqp;
                #pragma unroll
                for (int s = 0; s < S; ++s) sDot[s * N + wave] = dv[s];
            }
        }
        __syncthreads();

        // ---- Phase C: softmax over N + merge scalars (one thread per s) ----
        if (tid < S) {
            const int s = tid;
            const float rms2  = sqrtf(sSSP[s] * (1.0f / D) + EPSI);
            const float intra = sDQP[s] * SCALE / rms2;
            float lg[N];
            float imax = -INFINITY;
            #pragma unroll
            for (int n = 0; n < N; ++n) {
                lg[n] = sDot[s * N + n] * SCALE / sqrtf(sSSV[n] * (1.0f / D) + EPSI);
                imax = fmaxf(imax, lg[n]);
            }
            float sum = 0.f;
            #pragma unroll
            for (int n = 0; n < N; ++n) { lg[n] = expf(lg[n] - imax); sum += lg[n]; }
            const float mm   = fmaxf(imax, intra);
            const float wi   = expf(imax - mm);
            const float wp   = expf(intra - mm);
            const float nrm  = wi + wp;
            const float mlse = logf(wi * sum + wp) + mm;
            const float ca   = wi / (sum * nrm);
            #pragma unroll
            for (int n = 0; n < N; ++n) sCA[s * N + n] = lg[n] * ca;
            sCP[s] = wp / nrm;
            const long oidx = ((long)s * B + b) * T + t;
            outMax[oidx] = mm;
            outLse[oidx] = mlse;
        }
        __syncthreads();

        // ---- Phase D: merged_out = coefA(8x8) x V(8xD) + coefP * P, via f32 WMMA ----
        {
            const int m = lane & 15;
            const int h = lane >> 4;
            // A-matrix 16x4 layout: lanes 0-15 K={0,1}, lanes 16-31 K={2,3}; rows m>=8 zero pad
            v2f a0 = {0.f, 0.f}, a1 = {0.f, 0.f};
            if (m < S) {
                a0.x = sCA[m * N + 2*h];     a0.y = sCA[m * N + 2*h + 1];
                a1.x = sCA[m * N + 4 + 2*h]; a1.y = sCA[m * N + 5 + 2*h];
            }
            for (int dt = wave; dt < D / 16; dt += 8) {
                const int dcol = dt * 16 + m;
                // B-matrix 4x16: V0 = row(2h), V1 = row(2h+1); columns striped across lanes
                v2f b0, b1;
                b0.x = sV[(2*h    ) * VP + dcol];
                b0.y = sV[(2*h + 1) * VP + dcol];
                b1.x = sV[(2*h + 4) * VP + dcol];
                b1.y = sV[(2*h + 5) * VP + dcol];
                v8f c = {0.f,0.f,0.f,0.f,0.f,0.f,0.f,0.f};
                c = __builtin_amdgcn_wmma_f32_16x16x4_f32(false, a0, false, b0, (short)0, c, false, false);
                c = __builtin_amdgcn_wmma_f32_16x16x4_f32(false, a1, false, b1, (short)0, c, false, false);
                if (lane < 16) {
                    const int dd = dt * 16 + lane;
                    // Issue all 8 P loads first (one clause, one wait), then fma+store.
                    float pv[S];
                    #pragma unroll
                    for (int r = 0; r < S; ++r) pv[r] = pRow[r][dd];
                    #pragma unroll
                    for (int r = 0; r < S; ++r)      // C VGPR r = row M=r for lanes 0-15
                        oRow[r][dd] = c[r] + sCP[r] * pv[r];
                }
            }
        }
    }
}

extern "C" void kernel_launch(void* const* d_in, const int* in_sizes, int n_in,
                              void* d_out, int out_size, void* d_ws, size_t ws_size,
                              hipStream_t stream) {
    const float* q    = (const float*)d_in[0];   // pseudo_queries [S,D]
    const float* vrep = (const float*)d_in[1];   // block_reps    [N,B,T,D]
    const float* ps   = (const float*)d_in[2];   // partial_sums  [S,B,T,D]
    const float* wn   = (const float*)d_in[3];   // norm_weight   [D]
    float* out = (float*)d_out;

    dim3 grid(B * (T / TT));   // 1024 workgroups
    dim3 block(256);           // 8 waves (wave32)
    hipLaunchKernelGGL(twophase_attn_kernel, grid, block, 0, stream, q, vrep, ps, wn, out);
}